// NeighborConsistencyLoss_4844723110161
// MI455X (gfx1250) — compile-verified
//
#include <hip/hip_runtime.h>
#include <math.h>

typedef __attribute__((ext_vector_type(2))) float v2f;
typedef __attribute__((ext_vector_type(8))) float v8f;

#define D_DIM      512
#define K_NBR      32
#define ROW_STRIDE (D_DIM + 4)   // +4 floats pad -> conflict-free LDS banks
#define N_ROWS     (K_NBR + 1)   // 32 neighbor rows + 1 center row
#define EPS_CS     1e-8f

// One block per sampled center. 4 waves stage rows into LDS (coalesced
// float4), wave 0 runs 256x v_wmma_f32_16x16x4_f32 (full fp32 precision)
// to form dot(center, neigh_k) for k=0..31, norms via VALU + shfl.
__global__ __launch_bounds__(128)
void ncl_main_kernel(const float* __restrict__ z,
                     const int*   __restrict__ knn,
                     const int*   __restrict__ samp,
                     float*       __restrict__ partials)
{
    extern __shared__ float smem[];

    const int s    = blockIdx.x;
    const int sidx = samp[s];
    const int wave = threadIdx.x >> 5;
    const int lane = threadIdx.x & 31;
    const int* knn_row = knn + (size_t)sidx * K_NBR;

    // ---- stage rows 0..31 = neighbors, row 32 = center into LDS ----
    for (int row = wave; row < N_ROWS; row += 4) {
        const int srcRow = (row < K_NBR) ? knn_row[row] : sidx;
        const float4* g = (const float4*)(z + (size_t)srcRow * D_DIM);
        float4* d = (float4*)(smem + row * ROW_STRIDE);
        for (int j = lane; j < D_DIM / 4; j += 32) d[j] = g[j];
    }
    __syncthreads();

    // ---- wave 0: WMMA dot products + VALU norms ----
    if (threadIdx.x < 32) {
        const int hi = lane >> 4;      // lane half selects K=0,1 vs K=2,3
        const int r  = lane & 15;      // row (for A) / column (for B/C)
        const float* n0 = smem + r * ROW_STRIDE;          // neighbors 0..15
        const float* n1 = smem + (r + 16) * ROW_STRIDE;   // neighbors 16..31
        const float* cr = smem + K_NBR * ROW_STRIDE;      // center row

        v8f d0 = {0.f,0.f,0.f,0.f,0.f,0.f,0.f,0.f};
        v8f d1 = {0.f,0.f,0.f,0.f,0.f,0.f,0.f,0.f};
        float nn0 = 0.f, nn1 = 0.f, cc = 0.f;

        #pragma unroll 4
        for (int c = 0; c < D_DIM; c += 4) {
            const int off = c + 2 * hi;
            // f32 16x4 A layout: VGPR0 = K0 (lanes0-15) / K2 (lanes16-31),
            //                    VGPR1 = K1 / K3. B (4x16) mirrors it.
            v2f a0 = { n0[off], n0[off + 1] };   // A tile: neighbors 0..15
            v2f a1 = { n1[off], n1[off + 1] };   // A tile: neighbors 16..31
            v2f bc = { cr[off], cr[off + 1] };   // B tile: center bcast cols

            d0 = __builtin_amdgcn_wmma_f32_16x16x4_f32(
                     false, a0, false, bc, (short)0, d0, false, false);
            d1 = __builtin_amdgcn_wmma_f32_16x16x4_f32(
                     false, a1, false, bc, (short)0, d1, false, false);

            nn0 = fmaf(a0.x, a0.x, fmaf(a0.y, a0.y, nn0));
            nn1 = fmaf(a1.x, a1.x, fmaf(a1.y, a1.y, nn1));
            cc  = fmaf(bc.x, bc.x, fmaf(bc.y, bc.y, cc));
        }

        // each lane half saw 2 of every 4 elements -> combine halves
        nn0 += __shfl_xor(nn0, 16, 32);   // lane r (& r+16): ||neigh r||^2
        nn1 += __shfl_xor(nn1, 16, 32);   // lane r: ||neigh 16+r||^2
        cc  += __shfl_xor(cc,  16, 32);   // all lanes: ||center||^2
        const float nc = sqrtf(cc);

        // C layout: VGPR j, lanes0-15 -> M=j, lanes16-31 -> M=j+8;
        // all N columns identical (B broadcast), so d0[j] = dot(neigh M).
        float sumcos = 0.f;
        #pragma unroll
        for (int j = 0; j < 8; ++j) {
            const int m  = j + 8 * hi;                 // neighbor index (0..15)
            const float q0 = __shfl(nn0, m, 32);       // ||neigh m||^2
            const float q1 = __shfl(nn1, m, 32);       // ||neigh 16+m||^2
            sumcos += d0[j] / fmaxf(nc * sqrtf(q0), EPS_CS);
            sumcos += d1[j] / fmaxf(nc * sqrtf(q1), EPS_CS);
        }
        // lanes hi=0 hold cos-sum over neighbors 0-7 & 16-23... combine halves:
        sumcos += __shfl_xor(sumcos, 16, 32);          // full sum over all 32

        if (lane == 0)
            partials[s] = 1.0f - sumcos * (1.0f / (float)K_NBR);
    }
}

// Deterministic single-block tree reduction: out = mean(partials).
__global__ __launch_bounds__(256)
void ncl_reduce_kernel(const float* __restrict__ partials,
                       float* __restrict__ out, int S)
{
    __shared__ float buf[256];
    float acc = 0.f;
    for (int i = threadIdx.x; i < S; i += 256) acc += partials[i];
    buf[threadIdx.x] = acc;
    __syncthreads();
    for (int step = 128; step > 0; step >>= 1) {
        if ((int)threadIdx.x < step) buf[threadIdx.x] += buf[threadIdx.x + step];
        __syncthreads();
    }
    if (threadIdx.x == 0) out[0] = buf[0] / (float)S;
}

extern "C" void kernel_launch(void* const* d_in, const int* in_sizes, int n_in,
                              void* d_out, int out_size, void* d_ws, size_t ws_size,
                              hipStream_t stream) {
    const float* z    = (const float*)d_in[0];   // [N, 512] fp32
    const int*   knn  = (const int*)d_in[1];     // [N, 32] int32
    const int*   samp = (const int*)d_in[2];     // [S] int32
    float* out        = (float*)d_out;           // scalar fp32
    const int S       = in_sizes[2];

    float* partials = (float*)d_ws;              // S floats of scratch

    const size_t lds_bytes = (size_t)N_ROWS * ROW_STRIDE * sizeof(float);
    ncl_main_kernel<<<S, 128, lds_bytes, stream>>>(z, knn, samp, partials);
    ncl_reduce_kernel<<<1, 256, 0, stream>>>(partials, out, S);
}